// sRT_Optimizer_77730318123475
// MI455X (gfx1250) — compile-verified
//
#include <hip/hip_runtime.h>
#include <math.h>

typedef float v2f __attribute__((ext_vector_type(2)));
typedef float v4f __attribute__((ext_vector_type(4)));
typedef float v8f __attribute__((ext_vector_type(8)));

// ---------------------------------------------------------------------------
// Kernel 1: build the 4x4 transform (row-major, 16 floats) in d_ws.
// Mirrors the reference _se3_exp + scale/T assembly. ~100 flops, one thread.
// ---------------------------------------------------------------------------
__global__ void build_trans_kernel(const float* __restrict__ rot_delta,
                                   const float* __restrict__ trans_delta,
                                   const float* __restrict__ scale,
                                   const float* __restrict__ Rin,   // 3x3 row-major
                                   const float* __restrict__ Tin,   // 3
                                   float* __restrict__ trans)       // 16 floats out
{
    if (threadIdx.x != 0 || blockIdx.x != 0) return;

    float rho[3], th[3];
    for (int i = 0; i < 3; ++i) { rho[i] = trans_delta[i]; th[i] = rot_delta[i]; }

    const float W[3][3] = {{0.f,   -th[2],  th[1]},
                           {th[2],  0.f,   -th[0]},
                           {-th[1], th[0],  0.f }};
    float W2[3][3];
    for (int i = 0; i < 3; ++i)
        for (int j = 0; j < 3; ++j) {
            float s = 0.f;
            for (int k = 0; k < 3; ++k) s += W[i][k] * W[k][j];
            W2[i][j] = s;
        }

    const float ang   = sqrtf(th[0]*th[0] + th[1]*th[1] + th[2]*th[2]);
    const bool  small = ang < 1e-5f;
    const float a  = small ? 1.f : ang;
    const float sa = sinf(a), ca = cosf(a);
    const float Af = small ? 1.f        : sa / a;
    const float Bf = small ? 0.5f       : (1.f - ca) / (a * a);
    const float Cf = small ? (1.f/6.f)  : (a - sa) / (a * a * a);

    float Rd[3][3], Vm[3][3];
    for (int i = 0; i < 3; ++i)
        for (int j = 0; j < 3; ++j) {
            const float I = (i == j) ? 1.f : 0.f;
            Rd[i][j] = I + Af * W[i][j] + Bf * W2[i][j];
            Vm[i][j] = I + Bf * W[i][j] + Cf * W2[i][j];
        }

    float dt[3];
    for (int i = 0; i < 3; ++i)
        dt[i] = Vm[i][0]*rho[0] + Vm[i][1]*rho[1] + Vm[i][2]*rho[2];

    // top = scale * (Rd @ R)  (scale broadcasts over the LAST axis -> columns)
    for (int i = 0; i < 3; ++i) {
        for (int j = 0; j < 3; ++j) {
            const float m = Rd[i][0]*Rin[0*3+j] + Rd[i][1]*Rin[1*3+j] + Rd[i][2]*Rin[2*3+j];
            trans[i*4 + j] = scale[j] * m;
        }
        trans[i*4 + 3] = Tin[i] + dt[i];
    }
    trans[12] = 0.f; trans[13] = 0.f; trans[14] = 0.f; trans[15] = 1.f;
}

// ---------------------------------------------------------------------------
// Kernel 2: out.T = trans @ x.T via V_WMMA_F32_16X16X4_F32, 16 points/WMMA.
//   A (16x4) = trans (rows 4..15 zero), resident in 2 VGPRs for the whole wave
//   B (4x16) = x.T tile: lane n loads x[n][0:2], lane n+16 loads x[n][2:4]
//              -> one coalesced global_load_b64 (256 B / wave)
//   D rows 0..3 = out.T: lane n holds out[n][0..3] in d[0..3]
//              -> one contiguous global_store_b128 (lanes 0..15, 256 B)
//
// Fast path: all TILES_PER_WAVE loads issued back-to-back (constant 256 B
// immediate offsets) so the memory latency of tiles 1..3 hides behind the
// first WMMA instead of serializing load->wait->wmma per tile.
// ---------------------------------------------------------------------------
#define TILES_PER_WAVE 4

__global__ __launch_bounds__(256) void se3_transform_wmma_kernel(
    const float* __restrict__ x,      // (npoints, 4)
    const float* __restrict__ trans,  // 16 floats row-major
    float* __restrict__ out,          // (npoints, 4)
    unsigned npoints)
{
    const int lane = (int)(threadIdx.x & 31u);
    const int half = lane >> 4;    // 0: cols 0,1   1: cols 2,3
    const int n    = lane & 15;    // point-in-tile / matrix row

    // ---- A matrix (trans), ISA 16x4 f32 layout ----
    v2f aval = (v2f){0.f, 0.f};
    if (n < 4) {
        const float* p = trans + n * 4 + half * 2;
        aval.x = p[0];
        aval.y = p[1];
    }

    const unsigned wave = (blockIdx.x * blockDim.x + threadIdx.x) >> 5;
    const unsigned base = wave * (16u * TILES_PER_WAVE);

    // Per-lane base pointers; tile u is a constant +32 v2f (+256 B) offset.
    const v2f* __restrict__ ld = (const v2f*)x + ((size_t)(base + (unsigned)n) * 2u + (unsigned)half);
    v4f* __restrict__       st = (v4f*)out + (size_t)(base + (unsigned)n);

    if (base + 16u * TILES_PER_WAVE <= npoints) {
        // -------- fast path: loads first (MLP), then WMMAs + stores --------
        v2f b[TILES_PER_WAVE];
#pragma unroll
        for (int u = 0; u < TILES_PER_WAVE; ++u)
            b[u] = __builtin_nontemporal_load(ld + u * 32);

        v8f d[TILES_PER_WAVE];
#pragma unroll
        for (int u = 0; u < TILES_PER_WAVE; ++u) {
            v8f c = {};
            d[u] = __builtin_amdgcn_wmma_f32_16x16x4_f32(
                false, aval, false, b[u], (short)0, c, false, false);
        }

        if (half == 0) {
#pragma unroll
            for (int u = 0; u < TILES_PER_WAVE; ++u) {
                v4f o;
                o.x = d[u][0]; o.y = d[u][1]; o.z = d[u][2]; o.w = d[u][3];
                __builtin_nontemporal_store(o, st + u * 16);
            }
        }
    } else {
        // -------- tail path: per-tile guard (last partial block only) ------
#pragma unroll
        for (int u = 0; u < TILES_PER_WAVE; ++u) {
            const unsigned p0 = base + 16u * (unsigned)u;
            if (p0 + 16u <= npoints) {
                v2f bv = __builtin_nontemporal_load(ld + u * 32);
                v8f c = {};
                v8f dv = __builtin_amdgcn_wmma_f32_16x16x4_f32(
                    false, aval, false, bv, (short)0, c, false, false);
                if (half == 0) {
                    v4f o;
                    o.x = dv[0]; o.y = dv[1]; o.z = dv[2]; o.w = dv[3];
                    __builtin_nontemporal_store(o, st + u * 16);
                }
            }
        }
    }
}

// ---------------------------------------------------------------------------
// Launch. Input order: x, rot_delta, trans_delta, scale, R, T.
// ---------------------------------------------------------------------------
extern "C" void kernel_launch(void* const* d_in, const int* in_sizes, int n_in,
                              void* d_out, int out_size, void* d_ws, size_t ws_size,
                              hipStream_t stream)
{
    const float* x           = (const float*)d_in[0];
    const float* rot_delta   = (const float*)d_in[1];
    const float* trans_delta = (const float*)d_in[2];
    const float* scale       = (const float*)d_in[3];
    const float* R           = (const float*)d_in[4];
    const float* T           = (const float*)d_in[5];
    float*       out         = (float*)d_out;
    float*       trans       = (float*)d_ws;   // 16 floats of scratch

    const unsigned npoints = (unsigned)(in_sizes[0] / 4);

    build_trans_kernel<<<1, 32, 0, stream>>>(rot_delta, trans_delta, scale, R, T, trans);

    // 256 threads = 8 waves; each wave handles 16 * TILES_PER_WAVE = 64 points.
    const unsigned points_per_block = 8u * 16u * TILES_PER_WAVE;   // 512
    const unsigned blocks = (npoints + points_per_block - 1) / points_per_block;

    se3_transform_wmma_kernel<<<blocks, 256, 0, stream>>>(x, trans, out, npoints);
}